// InstanceSegmentationLoss_3221225472714
// MI455X (gfx1250) — compile-verified
//
#include <hip/hip_runtime.h>
#include <hip/hip_bf16.h>
#include <stdint.h>

// ---------------- problem constants (match reference setup_inputs) ---------
#define B_      2
#define HH      512
#define WW      512
#define PP      (HH*WW)          // 262144 pixels per image
#define KMAX    64               // static segment bound (jnp.unique size=64)
#define LAM     300.0f
#define BIGKEY  16777216u        // 256^3, pad value for unique keys
#define HASHN   256
#define SENT    0xFFFFFFFFu

// ---------------- workspace layout: WSB floats per batch -------------------
#define WSB      2048
#define OFF_HASH 0      // 256 x u32 open-addressed key set
#define OFF_UNIQ 256    // 64 x u32 sorted unique keys (padded BIGKEY)
#define OFF_CNT  320    // 64 f  per-segment pixel counts
#define OFF_SUMP 384    // 192 f per-segment sum of prediction (3ch)
#define OFF_SUMT 576    // 192 f per-segment sum of target (3ch)
#define OFF_HUB  768    // 64 f  per-segment huber sums
#define OFF_FSUM 832    // 64 f  sum_p f[p,k]
#define OFF_OWN  896    // 64 f  sum of f over own pixels
#define OFF_BM   960    // 256 f WMMA B-matrix entries: -2mx,-2my,-2mz,|m|^2
#define OFF_M3   1216   // 192 f per-segment prediction means
#define OFF_CTD  1408   // 64 f  counted flags
#define OFF_BG   1472   // 64 f  is-background flags
#define OFF_SCAL 1536   // [0]=mean_sep term, [1]=ct

typedef float v2f __attribute__((ext_vector_type(2)));
typedef float v8f __attribute__((ext_vector_type(8)));

// ---------------------------------------------------------------------------
__global__ void k_init(float* ws, float* out) {
    int i = blockIdx.x * blockDim.x + threadIdx.x;
    if (i == 0) out[0] = 0.0f;
    if (i < B_ * WSB) {
        int local = i % WSB;
        if (local < OFF_UNIQ) ((uint32_t*)ws)[i] = SENT;   // hash slots
        else                  ws[i] = 0.0f;                // accumulators
    }
}

// Gather the set of distinct target color keys per image.
__global__ void k_collect(const float* __restrict__ tgt, float* ws) {
    int gid = blockIdx.x * blockDim.x + threadIdx.x;
    if (gid >= B_ * PP) return;
    int b = gid / PP, p = gid % PP;
    const float* T = tgt + (size_t)b * 3 * PP;
    uint32_t r = (uint32_t)T[p], g = (uint32_t)T[PP + p], bl = (uint32_t)T[2 * PP + p];
    uint32_t key = r * 65536u + g * 256u + bl;           // exact (< 2^24)
    uint32_t* table = (uint32_t*)(ws + (size_t)b * WSB + OFF_HASH);
    uint32_t h = (key * 2654435761u) >> 24;              // 0..255
    for (int i = 0; i < HASHN; ++i) {
        uint32_t s = (h + i) & (HASHN - 1);
        uint32_t prev = atomicCAS(&table[s], SENT, key);
        if (prev == SENT || prev == key) break;
    }
}

// Compact the hash set into a sorted, BIGKEY-padded 64-entry table (== jnp.unique).
__global__ void k_sortuniq(float* ws) {
    int b = blockIdx.x;
    if (threadIdx.x != 0) return;
    uint32_t* table = (uint32_t*)(ws + (size_t)b * WSB + OFF_HASH);
    uint32_t* uniq  = (uint32_t*)(ws + (size_t)b * WSB + OFF_UNIQ);
    uint32_t tmp[KMAX];
    int cnt = 0;
    for (int i = 0; i < HASHN; ++i) {
        uint32_t k = table[i];
        if (k != SENT && cnt < KMAX) tmp[cnt++] = k;
    }
    for (int i = cnt; i < KMAX; ++i) tmp[i] = BIGKEY;
    for (int i = 1; i < KMAX; ++i) {             // insertion sort (<=64 elems)
        uint32_t v = tmp[i]; int j = i - 1;
        while (j >= 0 && tmp[j] > v) { tmp[j + 1] = tmp[j]; --j; }
        tmp[j + 1] = v;
    }
    for (int i = 0; i < KMAX; ++i) uniq[i] = tmp[i];
}

__device__ __forceinline__ int seg_search(const uint32_t* uq, uint32_t key) {
    int lo = 0, hi = KMAX;                        // lower_bound == searchsorted
    while (lo < hi) { int m = (lo + hi) >> 1; if (uq[m] < key) lo = m + 1; else hi = m; }
    return lo < KMAX ? lo : KMAX - 1;
}

// Per-segment counts / sum(pred) / sum(tgt) via LDS segmented reduction.
__global__ void k_reduce(const float* __restrict__ pred, const float* __restrict__ tgt,
                         float* ws) {
    __shared__ uint32_t uq[KMAX];
    __shared__ float cnt_s[KMAX], sp_s[KMAX * 3], st_s[KMAX * 3];
    int gid = blockIdx.x * blockDim.x + threadIdx.x;    // grid == B*P exactly
    int tid = threadIdx.x;
    int b = gid / PP, p = gid % PP;                     // whole block shares b
    float* base = ws + (size_t)b * WSB;
    if (tid < KMAX) { uq[tid] = ((uint32_t*)(base + OFF_UNIQ))[tid]; cnt_s[tid] = 0.0f; }
    if (tid < KMAX * 3) { sp_s[tid] = 0.0f; st_s[tid] = 0.0f; }
    __syncthreads();

    const float* T  = tgt  + (size_t)b * 3 * PP;
    const float* Pd = pred + (size_t)b * 3 * PP;
    float t0 = T[p], t1 = T[PP + p], t2 = T[2 * PP + p];
    uint32_t key = (uint32_t)t0 * 65536u + (uint32_t)t1 * 256u + (uint32_t)t2;
    int sid = seg_search(uq, key);
    atomicAdd(&cnt_s[sid], 1.0f);
    atomicAdd(&sp_s[sid * 3 + 0], Pd[p]);
    atomicAdd(&sp_s[sid * 3 + 1], Pd[PP + p]);
    atomicAdd(&sp_s[sid * 3 + 2], Pd[2 * PP + p]);
    atomicAdd(&st_s[sid * 3 + 0], t0);
    atomicAdd(&st_s[sid * 3 + 1], t1);
    atomicAdd(&st_s[sid * 3 + 2], t2);
    __syncthreads();
    if (tid < KMAX) atomicAdd(base + OFF_CNT + tid, cnt_s[tid]);
    if (tid < KMAX * 3) {
        atomicAdd(base + OFF_SUMP + tid, sp_s[tid]);
        atomicAdd(base + OFF_SUMT + tid, st_s[tid]);
    }
}

// Means, bg/counted flags, WMMA B-matrix, pairwise-mean repulsion term.
__global__ void k_seg(float* ws, const unsigned char* __restrict__ no_bg) {
    int b = blockIdx.x, k = threadIdx.x;                // 64 threads
    float* base = ws + (size_t)b * WSB;
    __shared__ float red[KMAX];
    __shared__ float me_s[KMAX * 3];
    __shared__ float ctd_s[KMAX];

    float cnt = base[OFF_CNT + k];
    float n   = fmaxf(cnt, 1.0f);
    float mx = base[OFF_SUMP + 3 * k + 0] / n;
    float my = base[OFF_SUMP + 3 * k + 1] / n;
    float mz = base[OFF_SUMP + 3 * k + 2] / n;
    float tsum = (base[OFF_SUMT + 3 * k + 0] + base[OFF_SUMT + 3 * k + 1] +
                  base[OFF_SUMT + 3 * k + 2]) / n;
    int isbg  = fabsf(tsum) < 1e-5f;
    int valid = cnt > 0.0f;
    int nb    = no_bg[b] != 0;
    int counted = valid && (!isbg || !nb);

    base[OFF_BM + 4 * k + 0] = -2.0f * mx;              // A row dot B col =
    base[OFF_BM + 4 * k + 1] = -2.0f * my;              //   -2 p.m + |m|^2
    base[OFF_BM + 4 * k + 2] = -2.0f * mz;
    base[OFF_BM + 4 * k + 3] = mx * mx + my * my + mz * mz;
    base[OFF_M3 + 3 * k + 0] = mx;
    base[OFF_M3 + 3 * k + 1] = my;
    base[OFF_M3 + 3 * k + 2] = mz;
    base[OFF_CTD + k] = counted ? 1.0f : 0.0f;
    base[OFF_BG  + k] = isbg ? 1.0f : 0.0f;

    me_s[3 * k + 0] = isbg ? 0.0f : mx;                 // bg mean zeroed for pairs
    me_s[3 * k + 1] = isbg ? 0.0f : my;
    me_s[3 * k + 2] = isbg ? 0.0f : mz;
    ctd_s[k] = counted ? 1.0f : 0.0f;
    red[k]   = ctd_s[k];
    __syncthreads();
    for (int s = 32; s > 0; s >>= 1) { if (k < s) red[k] += red[k + s]; __syncthreads(); }
    float ct = red[0];
    __syncthreads();

    float ps = 0.0f;
    if (ctd_s[k] > 0.0f) {
        for (int j = 0; j < KMAX; ++j) {
            if (j == k || ctd_s[j] == 0.0f) continue;
            float dx = me_s[3 * k] - me_s[3 * j];
            float dy = me_s[3 * k + 1] - me_s[3 * j + 1];
            float dz = me_s[3 * k + 2] - me_s[3 * j + 2];
            float sq = dx * dx + dy * dy + dz * dz;
            ps += LAM / (sq + 1.0f);                    // tiny K*K: keep exact
        }
    }
    red[k] = ps;
    __syncthreads();
    for (int s = 32; s > 0; s >>= 1) { if (k < s) red[k] += red[k + s]; __syncthreads(); }
    if (k == 0) {
        float pair_sum = red[0] * 0.5f;
        float npairs   = ct * (ct - 1.0f) * 0.5f;
        base[OFF_SCAL + 0] = (ct > 1.0f) ? pair_sum / fmaxf(npairs, 1.0f) : 0.0f;
        base[OFF_SCAL + 1] = ct;
    }
}

// Main kernel: per-pixel Huber + the [P,64] separation matrix via 4 unrolled
// V_WMMA_F32_16X16X4_F32 (A: 16 pixels x [px,py,pz,1], B: [-2m,|m|^2] x 16 segs).
// f = LAM/(1+d) uses v_rcp_f32 (1-ulp approx) instead of IEEE div: the exact
// division expanded to ~7 VALU ops per element and dominated the 16.7M-element
// elementwise tail; the approximation error is negligible in the reduced sums.
__global__ void __launch_bounds__(256) k_pixel(const float* __restrict__ pred,
                                               const float* __restrict__ tgt,
                                               float* ws) {
    __shared__ float    bm_s[KMAX * 4];
    __shared__ float    m3_s[KMAX * 3];
    __shared__ float    bg_s[KMAX];
    __shared__ uint32_t uq_s[KMAX];
    __shared__ float    fs_s[KMAX], ow_s[KMAX], hb_s[KMAX];
    __shared__ float    pf2_s[8][16];
    __shared__ int      sid_s[8][16];

    int blocks_per_b = PP / 128;                        // 8 waves * 16 px = 128/block
    int b     = blockIdx.x / blocks_per_b;
    int chunk = blockIdx.x % blocks_per_b;
    int tid = threadIdx.x;
    int wave = tid >> 5, lane = tid & 31;
    int half = lane >> 4, m16 = lane & 15;
    float* base = ws + (size_t)b * WSB;

    if (tid < KMAX * 4) bm_s[tid] = base[OFF_BM + tid];
    if (tid < KMAX * 3) m3_s[tid] = base[OFF_M3 + tid];
    if (tid < KMAX) {
        bg_s[tid] = base[OFF_BG + tid];
        uq_s[tid] = ((uint32_t*)(base + OFF_UNIQ))[tid];
        fs_s[tid] = 0.0f; ow_s[tid] = 0.0f; hb_s[tid] = 0.0f;
    }
    __syncthreads();

    const float* Pd = pred + (size_t)b * 3 * PP;
    const float* T  = tgt  + (size_t)b * 3 * PP;
    int myp = chunk * 128 + wave * 16 + m16;            // pixel M = lane&15
    float px = Pd[myp], py = Pd[PP + myp], pz = Pd[2 * PP + myp];

    if (half == 0) {                                    // lanes 0-15: stage pf2/sid, huber
        pf2_s[wave][m16] = px * px + py * py + pz * pz;
        uint32_t key = (uint32_t)T[myp] * 65536u + (uint32_t)T[PP + myp] * 256u +
                       (uint32_t)T[2 * PP + myp];
        int sid = seg_search(uq_s, key);
        sid_s[wave][m16] = sid;
        float rx = 0.0f, ry = 0.0f, rz = 0.0f;
        if (bg_s[sid] == 0.0f) { rx = m3_s[3 * sid]; ry = m3_s[3 * sid + 1]; rz = m3_s[3 * sid + 2]; }
        float h = 0.0f;
        { float e = px - rx, ae = fabsf(e); h += (ae < 1.0f) ? 0.5f * e * e : ae - 0.5f; }
        { float e = py - ry, ae = fabsf(e); h += (ae < 1.0f) ? 0.5f * e * e : ae - 0.5f; }
        { float e = pz - rz, ae = fabsf(e); h += (ae < 1.0f) ? 0.5f * e * e : ae - 0.5f; }
        atomicAdd(&hb_s[sid], h);
    }
    __syncthreads();

    // A operand, 16x4 f32: lane holds M=lane&15, K = 2*half + vgpr.
    v2f a;
    a.x = half ? pz : px;
    a.y = half ? 1.0f : py;

    // Hoist all four B tiles, then issue 4 independent back-to-back WMMAs.
    v2f bv[4];
#pragma unroll
    for (int t = 0; t < 4; ++t) {
        int seg = t * 16 + m16;                         // B: N=lane&15, K=2*half+vgpr
        bv[t].x = bm_s[seg * 4 + 2 * half + 0];
        bv[t].y = bm_s[seg * 4 + 2 * half + 1];
    }
    v8f c[4];
#pragma unroll
    for (int t = 0; t < 4; ++t) {
        v8f cz = {};
        c[t] = __builtin_amdgcn_wmma_f32_16x16x4_f32(
            /*neg_a=*/false, a, /*neg_b=*/false, bv[t],
            /*c_mod=*/(short)0, cz, /*reuse_a=*/false, /*reuse_b=*/false);
    }

    // Stage this wave's pf2/sid rows into registers once.
    float pf2r[8];
    int   sidr[8];
#pragma unroll
    for (int j = 0; j < 8; ++j) {
        int M = half * 8 + j;                           // C: lane holds M = 8*half+j
        pf2r[j] = pf2_s[wave][M];
        sidr[j] = sid_s[wave][M];
    }

#pragma unroll
    for (int t = 0; t < 4; ++t) {
        int seg = t * 16 + m16;
        float accf = 0.0f, acco = 0.0f;
#pragma unroll
        for (int j = 0; j < 8; ++j) {
            float d = fmaxf(pf2r[j] + c[t][j], 0.0f);
            float f = LAM * __builtin_amdgcn_rcpf(1.0f + d);   // v_rcp_f32
            accf += f;
            if (sidr[j] == seg) acco += f;
        }
        atomicAdd(&fs_s[seg], accf);
        atomicAdd(&ow_s[seg], acco);
    }
    __syncthreads();
    if (tid < KMAX) {
        atomicAdd(base + OFF_FSUM + tid, fs_s[tid]);
        atomicAdd(base + OFF_OWN  + tid, ow_s[tid]);
        atomicAdd(base + OFF_HUB  + tid, hb_s[tid]);
    }
}

__global__ void k_final(float* ws, float* out) {
    int b = blockIdx.x, k = threadIdx.x;                // 64 threads
    float* base = ws + (size_t)b * WSB;
    __shared__ float red[KMAX];
    float cnt = base[OFF_CNT + k];
    float n = fmaxf(cnt, 1.0f);
    float counted = base[OFF_CTD + k];
    float isbg    = base[OFF_BG + k];
    float hub = (counted > 0.0f) ? base[OFF_HUB + k] / (3.0f * n) : 0.0f;
    float n_out = (float)PP - cnt;
    float sep_mean = (base[OFF_FSUM + k] - base[OFF_OWN + k]) / fmaxf(n_out, 1.0f);
    int sep_active = (cnt > 0.0f) && (isbg == 0.0f) && (n_out > 0.0f);
    float sep = sep_active ? (10.0f / sqrtf(n)) * sep_mean : 0.0f;
    red[k] = hub + sep;
    __syncthreads();
    for (int s = 32; s > 0; s >>= 1) { if (k < s) red[k] += red[k + s]; __syncthreads(); }
    if (k == 0) {
        float ct = base[OFF_SCAL + 1];
        float loss = (red[0] + base[OFF_SCAL + 0]) / fmaxf(ct, 1.0f);
        atomicAdd(out, loss / (float)B_);
    }
}

// ---------------------------------------------------------------------------
extern "C" void kernel_launch(void* const* d_in, const int* in_sizes, int n_in,
                              void* d_out, int out_size, void* d_ws, size_t ws_size,
                              hipStream_t stream) {
    const float* pred = (const float*)d_in[0];          // [B,3,H,W] f32
    const float* tgt  = (const float*)d_in[1];          // [B,3,H,W] f32
    const unsigned char* nobg = (const unsigned char*)d_in[2]; // bool[2]
    float* out = (float*)d_out;
    float* ws  = (float*)d_ws;                          // needs B_*WSB*4 = 16 KB

    k_init   <<<(B_ * WSB + 255) / 256, 256, 0, stream>>>(ws, out);
    k_collect<<<(B_ * PP + 255) / 256, 256, 0, stream>>>(tgt, ws);
    k_sortuniq<<<B_, 32, 0, stream>>>(ws);
    k_reduce <<<B_ * PP / 256, 256, 0, stream>>>(pred, tgt, ws);
    k_seg    <<<B_, 64, 0, stream>>>(ws, nobg);
    k_pixel  <<<B_ * PP / 128, 256, 0, stream>>>(pred, tgt, ws);
    k_final  <<<B_, 64, 0, stream>>>(ws, out);
    (void)in_sizes; (void)n_in; (void)out_size; (void)ws_size;
}